// MaskedFocalLoss_50629074485708
// MI455X (gfx1250) — compile-verified
//
#include <hip/hip_runtime.h>
#include <hip/hip_bf16.h>
#include <cmath>

// ---------------------------------------------------------------------------
// MaskedFocalLoss for MI455X (gfx1250, wave32).
// Phase 0: zero workspace (uint4 stores, ~84MB @ 23.3 TB/s ~ 3.6us)
// Phase 1: pixel histograms, int4-vectorized stream (134MB @ 23.3 TB/s ~ 5.7us).
//          LDS-privatized n_p/n_t (42KB LDS); pair atomics direct to L2
//          (78.6MB pair array fits in the 192MB L2).
// Phase 2: uint4 scan of 19.66M pair bins -> IoU weights -> out[8192,150].
// Phase 3: per-row focal loss; the three per-row sums run on
//          V_WMMA_F32_16X16X4_F32 (16 rows per wave32, B = all-ones),
//          37 branch-free K-chunks + scalar tail for cols 148..149.
// ---------------------------------------------------------------------------

#define N_CLASSES 150
#define N_PRED    8192
#define BATCH     16
#define TDIM      (BATCH * N_CLASSES)       // 2400
#define NPIX      (BATCH * 1024 * 1024)     // 16,777,216 (H*W = 2^20)
#define NBINS     (N_PRED * TDIM)           // 19,660,800

typedef float v2f __attribute__((ext_vector_type(2)));
typedef float v8f __attribute__((ext_vector_type(8)));

// ---------------- Phase 0: zero init ----------------
__global__ __launch_bounds__(256) void mfl_zero(uint4* __restrict__ w, long long nvec,
                                                float* __restrict__ loss) {
  long long i = (long long)blockIdx.x * blockDim.x + threadIdx.x;
  long long stride = (long long)gridDim.x * blockDim.x;
  uint4 z = make_uint4(0u, 0u, 0u, 0u);
  for (; i < nvec; i += stride) w[i] = z;
  if (blockIdx.x == 0 && threadIdx.x == 0) *loss = 0.0f;
}

// ---------------- Phase 1: histograms (int4 stream) ----------------
__global__ __launch_bounds__(256) void mfl_hist(const int4* __restrict__ predseg4,
                                                const int4* __restrict__ targetseg4,
                                                unsigned* __restrict__ n_ov,
                                                unsigned* __restrict__ n_p,
                                                unsigned* __restrict__ n_t) {
  __shared__ unsigned s_np[N_PRED];   // 32 KB
  __shared__ unsigned s_nt[TDIM];     // 9.6 KB
  for (int i = threadIdx.x; i < N_PRED; i += blockDim.x) s_np[i] = 0u;
  for (int i = threadIdx.x; i < TDIM;   i += blockDim.x) s_nt[i] = 0u;
  __syncthreads();

  const int nquads = NPIX / 4;        // 4,194,304
  int stride = blockDim.x * gridDim.x;
  for (int i = blockIdx.x * blockDim.x + threadIdx.x; i < nquads; i += stride) {
    int4 p4 = predseg4[i];
    int4 c4 = targetseg4[i];
    int tb = (i >> 18) * N_CLASSES;   // batch*(H*W)=2^20 px => quad 2^18; same b for all 4
    int t0 = tb + c4.x, t1 = tb + c4.y, t2 = tb + c4.z, t3 = tb + c4.w;
    atomicAdd(&s_np[p4.x], 1u);
    atomicAdd(&s_np[p4.y], 1u);
    atomicAdd(&s_np[p4.z], 1u);
    atomicAdd(&s_np[p4.w], 1u);
    atomicAdd(&s_nt[t0], 1u);
    atomicAdd(&s_nt[t1], 1u);
    atomicAdd(&s_nt[t2], 1u);
    atomicAdd(&s_nt[t3], 1u);
    atomicAdd(&n_ov[p4.x * TDIM + t0], 1u);
    atomicAdd(&n_ov[p4.y * TDIM + t1], 1u);
    atomicAdd(&n_ov[p4.z * TDIM + t2], 1u);
    atomicAdd(&n_ov[p4.w * TDIM + t3], 1u);
  }
  __syncthreads();
  for (int i = threadIdx.x; i < N_PRED; i += blockDim.x) {
    unsigned v = s_np[i];
    if (v) atomicAdd(&n_p[i], v);
  }
  for (int i = threadIdx.x; i < TDIM; i += blockDim.x) {
    unsigned v = s_nt[i];
    if (v) atomicAdd(&n_t[i], v);
  }
}

// ---------------- Phase 2: pair bins -> out[p, cls] (uint4 scan) ------------
__global__ __launch_bounds__(256) void mfl_pairs(const uint4* __restrict__ n_ov4,
                                                 const unsigned* __restrict__ n_p,
                                                 const unsigned* __restrict__ n_t,
                                                 float* __restrict__ out) {
  const int nquads = NBINS / 4;       // 4,915,200 ; TDIM%4==0 => one p per quad
  int stride = blockDim.x * gridDim.x;
  for (int q = blockIdx.x * blockDim.x + threadIdx.x; q < nquads; q += stride) {
    uint4 c = n_ov4[q];
    if ((c.x | c.y | c.z | c.w) == 0u) continue;
    int j = q * 4;
    int p = j / TDIM;
    int t = j - p * TDIM;             // t..t+3 share this p
    unsigned np = n_p[p];
    float* orow = out + p * N_CLASSES;
    if (c.x) { int tt = t;     atomicAdd(&orow[tt % N_CLASSES], (float)c.x / (float)(np + n_t[tt] - c.x)); }
    if (c.y) { int tt = t + 1; atomicAdd(&orow[tt % N_CLASSES], (float)c.y / (float)(np + n_t[tt] - c.y)); }
    if (c.z) { int tt = t + 2; atomicAdd(&orow[tt % N_CLASSES], (float)c.z / (float)(np + n_t[tt] - c.z)); }
    if (c.w) { int tt = t + 3; atomicAdd(&orow[tt % N_CLASSES], (float)c.w / (float)(np + n_t[tt] - c.w)); }
  }
}

// ---------------- Phase 3 helpers ----------------
// Extract per-row scalar from the 16x16 f32 C/D accumulator (B = all-ones so
// every column of D holds the row-sum). VGPR r holds M=r (lanes 0-15) or
// M=r+8 (lanes 16-31); lane pair (l, l+16) both work on row m = l&15.
__device__ __forceinline__ float mfl_extract_row(v8f d, int lane) {
  int r = lane & 7;
  float v01 = (r & 1) ? d[1] : d[0];
  float v23 = (r & 1) ? d[3] : d[2];
  float v45 = (r & 1) ? d[5] : d[4];
  float v67 = (r & 1) ? d[7] : d[6];
  float v03 = (r & 2) ? v23 : v01;
  float v47 = (r & 2) ? v67 : v45;
  float own = (r & 4) ? v47 : v03;
  float other = __shfl_xor(own, 16, 32);
  bool own_valid = (((lane & 15) < 8) == (lane < 16));
  return own_valid ? own : other;
}

// broadcast a value computed by lanes 0..15 to the partner lane (+16)
__device__ __forceinline__ float mfl_bcast_lo(float v, int lane) {
  float o = __shfl_xor(v, 16, 32);
  return (lane < 16) ? v : o;
}

// ---------------- Phase 3: per-row focal loss (16 rows per wave32) ----------
__global__ __launch_bounds__(32) void mfl_loss(const float* __restrict__ pred,
                                               const float* __restrict__ out,
                                               float* __restrict__ loss_out,
                                               float cnorm) {
  const int lane = threadIdx.x;                 // 0..31, one full wave
  const int m    = lane & 15;
  const int row  = blockIdx.x * 16 + m;
  const float* prow = pred + (long long)row * N_CLASSES;
  const float* orow = out  + (long long)row * N_CLASSES;
  const int cBase = (lane >= 16) ? 2 : 0;       // A-matrix K layout (ISA 7.12.2)

  v2f ones; ones.x = 1.0f; ones.y = 1.0f;       // B = all-ones 4x16

  // ---- loop 1: S1 = rowsum(out) via WMMA; fused max(pred) + argmax(out) ----
  // 37 branch-free chunks cover cols 0..147 for every lane.
  v8f accS1 = {};
  float pmax  = -3.402823466e38f;
  float obest = -3.402823466e38f;
  int   oidx  = 0;
  for (int k = 0; k < 37; ++k) {
    int c0 = 4 * k + cBase;
    float o0 = orow[c0];
    float o1 = orow[c0 + 1];
    float p0 = prow[c0];
    float p1 = prow[c0 + 1];
    pmax = fmaxf(pmax, fmaxf(p0, p1));
    if (o0 > obest) { obest = o0; oidx = c0; }
    if (o1 > obest) { obest = o1; oidx = c0 + 1; }
    v2f a; a.x = o0; a.y = o1;
    accS1 = __builtin_amdgcn_wmma_f32_16x16x4_f32(false, a, false, ones,
                                                  (short)0, accS1, false, false);
  }
  // tail cols 148,149 handled by lanes 0..15 (single divergent region, no wmma)
  float tail1 = 0.0f;
  if (lane < 16) {
    float o148 = orow[148], o149 = orow[149];
    float p148 = prow[148], p149 = prow[149];
    pmax = fmaxf(pmax, fmaxf(p148, p149));
    if (o148 > obest) { obest = o148; oidx = 148; }
    if (o149 > obest) { obest = o149; oidx = 149; }
    tail1 = o148 + o149;
  }
  // combine lane pair: row max, first-index argmax (jnp.argmax tie-break)
  pmax = fmaxf(pmax, __shfl_xor(pmax, 16, 32));
  {
    float ob2 = __shfl_xor(obest, 16, 32);
    int   oi2 = __shfl_xor(oidx, 16, 32);
    if (ob2 > obest || (ob2 == obest && oi2 < oidx)) { obest = ob2; oidx = oi2; }
  }
  float S1 = mfl_extract_row(accS1, lane) + mfl_bcast_lo(tail1, lane);

  // ---- loop 2: S2 = sum exp(pred - max) via WMMA ----
  v8f accS2 = {};
  for (int k = 0; k < 37; ++k) {
    int c0 = 4 * k + cBase;
    v2f a;
    a.x = expf(prow[c0]     - pmax);
    a.y = expf(prow[c0 + 1] - pmax);
    accS2 = __builtin_amdgcn_wmma_f32_16x16x4_f32(false, a, false, ones,
                                                  (short)0, accS2, false, false);
  }
  float tail2 = 0.0f;
  if (lane < 16) {
    tail2 = expf(prow[148] - pmax) + expf(prow[149] - pmax);
  }
  float S2 = mfl_extract_row(accS2, lane) + mfl_bcast_lo(tail2, lane);
  float logZ = logf(S2);

  // ---- loop 3: S3 = sum_c out[c]*logp[c] via WMMA (all c, fix c=0 after) ----
  v8f accS3 = {};
  for (int k = 0; k < 37; ++k) {
    int c0 = 4 * k + cBase;
    v2f a;
    a.x = orow[c0]     * (prow[c0]     - pmax - logZ);
    a.y = orow[c0 + 1] * (prow[c0 + 1] - pmax - logZ);
    accS3 = __builtin_amdgcn_wmma_f32_16x16x4_f32(false, a, false, ones,
                                                  (short)0, accS3, false, false);
  }
  // tail cols 148,149 minus the class-0 term (cls_w[0] = 0)
  float adj3 = 0.0f;
  if (lane < 16) {
    float lp148 = prow[148] - pmax - logZ;
    float lp149 = prow[149] - pmax - logZ;
    float lp0   = prow[0]   - pmax - logZ;
    adj3 = orow[148] * lp148 + orow[149] * lp149 - orow[0] * lp0;
  }
  float S3 = mfl_extract_row(accS3, lane) + mfl_bcast_lo(adj3, lane);

  // ce = -sum(cls_w * (out/S1) * logp) = -S3/S1 ; pt = softmax[argmax]
  float ce = -S3 / S1;
  float pt = expf(prow[oidx] - pmax) / S2;
  float q  = 1.0f - pt;
  float loss = (q * q) * (q * q) * ce * cnorm;

  if (lane < 16) {
    atomicAdd(loss_out, loss * (1.0f / (float)N_PRED));
  }
}

// ---------------- Host: trapz norm factor (mirrors jnp float32 math) --------
static float mfl_norm_factor() {
  const float gamma = 4.0f, eps = 1e-7f;
  float h = 0.0f;
  float tprev = 0.0f, yprev = 1.0f;   // t=0 -> y = (1-0)/(1-0) = 1
  for (int i = 1; i < 1000; ++i) {
    float t = (float)i * (1.0f - eps) / 999.0f;
    float y = (1.0f - powf(t, gamma + 1.0f)) / (1.0f - t);
    h += 0.5f * (y + yprev) * (t - tprev);
    tprev = t; yprev = y;
  }
  return (gamma + 1.0f) / h;
}

extern "C" void kernel_launch(void* const* d_in, const int* in_sizes, int n_in,
                              void* d_out, int out_size, void* d_ws, size_t ws_size,
                              hipStream_t stream) {
  const float* pred      = (const float*)d_in[0];   // [8192, 150] f32
  const int*   predseg   = (const int*)d_in[1];     // [16,1024,1024] i32
  const int*   targetseg = (const int*)d_in[2];     // [16,1024,1024] i32
  float* loss_out = (float*)d_out;                  // scalar f32

  // Workspace layout (bytes), all 16B aligned:
  //   n_ov : 8192*2400*4 = 78,643,200
  //   n_p  : 8192*4      = 32,768
  //   n_t  : 2400*4      = 9,600
  //   out  : 8192*150*4  = 4,915,200
  char* ws = (char*)d_ws;
  unsigned* n_ov = (unsigned*)(ws);
  unsigned* n_p  = (unsigned*)(ws + 78643200);
  unsigned* n_t  = (unsigned*)(ws + 78675968);
  float*    outw = (float*)   (ws + 78685568);
  const long long total_words = 19660800LL + 8192 + 2400 + (8192LL * 150);
  const long long nvec = (total_words + 3) / 4;

  float cnorm = mfl_norm_factor();

  mfl_zero <<<2048, 256, 0, stream>>>((uint4*)ws, nvec, loss_out);
  mfl_hist <<<1024, 256, 0, stream>>>((const int4*)predseg, (const int4*)targetseg,
                                      n_ov, n_p, n_t);
  mfl_pairs<<<4096, 256, 0, stream>>>((const uint4*)n_ov, n_p, n_t, outw);
  mfl_loss <<<N_PRED / 16, 32, 0, stream>>>(pred, outw, loss_out, cnorm);
}